// WeightedSVDSolver_50611894616454
// MI455X (gfx1250) — compile-verified
//
#include <hip/hip_runtime.h>
#include <math.h>

typedef __attribute__((ext_vector_type(2))) float v2f;
typedef __attribute__((ext_vector_type(8))) float v8f;

#define L_DIM 768
#define M_DIM 768
#define TILE_ROWS 16
#define WAVES_PER_BLOCK 8
#define TILE_GROUPS (L_DIM / (TILE_ROWS * WAVES_PER_BLOCK)) /* 6 */
#define KCHUNKS (M_DIM / 4)                                 /* 192 */

// ---------------------------------------------------------------------------
// zero accumulators (d_ws is poisoned, must init every call)
// ---------------------------------------------------------------------------
__global__ void wsvd_zero(float* __restrict__ acc, int n) {
    int i = blockIdx.x * blockDim.x + threadIdx.x;
    if (i < n) acc[i] = 0.0f;
}

// ---------------------------------------------------------------------------
// Stage 1: fused  tgt_matched = corr @ tgt  (WMMA f32 16x16x4)  +  all
// row-reductions: per batch b accumulate into acc[b*16 + ...]:
//   [0..8]  sum_l w*src[c]*tgtm[d]   [9..11] sum_l w*tgtm
//   [12..14] sum_l w*src             [15]    sum_l w
// One wave = one 16-row tile; block = 8 waves of one batch slice.
// B fragments are pre-expanded into LDS in per-lane WMMA layout so the
// inner loop is branch-free: global_load_b64 + ds_load_b64 + v_wmma.
// ---------------------------------------------------------------------------
__global__ __launch_bounds__(256) void wsvd_stage1(
    const float* __restrict__ corr,     // [B, L, M]
    const float* __restrict__ overlap,  // [B, L]
    const float* __restrict__ src,      // [B, L, 3]
    const float* __restrict__ tgt,      // [B, M, 3]
    float* __restrict__ acc)            // [B, 16]
{
    __shared__ float sTgt[M_DIM * 3];                      //  9216 B
    __shared__ v2f   sB[KCHUNKS * 32];                     // 49152 B
    __shared__ float sTgtm[WAVES_PER_BLOCK][TILE_ROWS][4]; //  2048 B

    const int b    = blockIdx.x / TILE_GROUPS;
    const int g    = blockIdx.x % TILE_GROUPS;
    const int tid  = threadIdx.x;
    const int wave = tid >> 5;
    const int lane = tid & 31;

    // cooperative coalesced load of tgt[b] into LDS
    const float* tb = tgt + (size_t)b * (M_DIM * 3);
    for (int i = tid; i < M_DIM * 3; i += 256) sTgt[i] = tb[i];
    __syncthreads();

    // pre-expand B fragments: sB[c][j] = { tgt[c*4 + 2*kh][n], tgt[c*4+2*kh+1][n] }
    // for n = j&15 (<3 else 0), kh = j>>4.  One-time 48 KB; zeros baked in.
    for (int e = tid; e < KCHUNKS * 32; e += 256) {
        const int c  = e >> 5;
        const int j  = e & 31;
        const int n  = j & 15;
        const int kh = j >> 4;
        const int k0 = (c * 4 + kh * 2) * 3 + n;
        v2f v;
        if (n < 3) { v.x = sTgt[k0]; v.y = sTgt[k0 + 3]; }
        else       { v.x = 0.0f;     v.y = 0.0f; }
        sB[e] = v;
    }
    __syncthreads();

    const int tile    = g * WAVES_PER_BLOCK + wave;
    const int rowBase = tile * TILE_ROWS;
    const int n       = lane & 15;   // A-matrix row-within-tile / WMMA N index
    const int khalf   = lane >> 4;   // 0 -> K={0,1}, 1 -> K={2,3}

    // A fragment stream: lane holds row (rowBase+n), K pair (2*khalf, 2*khalf+1)
    const float* arow = corr + (size_t)b * L_DIM * M_DIM
                      + (size_t)(rowBase + n) * M_DIM + (khalf << 1);
    const v2f* bp = sB + lane;

    v8f c8 = {};
#pragma unroll 8
    for (int c = 0; c < KCHUNKS; ++c) {
        v2f a  = *(const v2f*)(arow + c * 4);
        v2f bf = bp[c * 32];
        c8 = __builtin_amdgcn_wmma_f32_16x16x4_f32(false, a, false, bf,
                                                   (short)0, c8, false, false);
    }

    // C layout: VGPR r, lanes 0-15 -> (row rowBase+r,   col lane),
    //                   lanes16-31 -> (row rowBase+8+r, col lane-16)
    if (n < 3) {
        const int r0 = khalf * 8;
#pragma unroll
        for (int r = 0; r < 8; ++r)
            sTgtm[wave][r0 + r][n] = c8[r];
    }
    asm volatile("s_wait_dscnt 0" ::: "memory");  // intra-wave LDS handoff

    float red[16];
#pragma unroll
    for (int i = 0; i < 16; ++i) red[i] = 0.0f;

    if (lane < 16) {
        const int l = rowBase + lane;
        const float w  = overlap[(size_t)b * L_DIM + l];
        const float* sp = src + ((size_t)b * L_DIM + l) * 3;
        const float s0 = sp[0], s1 = sp[1], s2 = sp[2];
        const float t0 = sTgtm[wave][lane][0];
        const float t1 = sTgtm[wave][lane][1];
        const float t2 = sTgtm[wave][lane][2];
        red[0] = w * s0 * t0; red[1] = w * s0 * t1; red[2] = w * s0 * t2;
        red[3] = w * s1 * t0; red[4] = w * s1 * t1; red[5] = w * s1 * t2;
        red[6] = w * s2 * t0; red[7] = w * s2 * t1; red[8] = w * s2 * t2;
        red[9]  = w * t0; red[10] = w * t1; red[11] = w * t2;
        red[12] = w * s0; red[13] = w * s1; red[14] = w * s2;
        red[15] = w;
    }

    // wave32 butterfly reduction
#pragma unroll
    for (int off = 16; off > 0; off >>= 1) {
#pragma unroll
        for (int i = 0; i < 16; ++i)
            red[i] += __shfl_xor(red[i], off, 32);
    }

    if (lane == 0) {
        float* ab = acc + b * 16;
#pragma unroll
        for (int i = 0; i < 16; ++i)
            atomicAdd(ab + i, red[i]);
    }
}

// ---------------------------------------------------------------------------
// Stage 2: per-batch 3x3 Kabsch (Jacobi eigen of H^T H), t, quaternion
// ---------------------------------------------------------------------------
__device__ __forceinline__ void jacobi_rot(float S[3][3], float V[3][3],
                                           int p, int q) {
    float apq = S[p][q];
    if (fabsf(apq) < 1e-20f) return;
    float tau = (S[q][q] - S[p][p]) / (2.0f * apq);
    float t   = ((tau >= 0.0f) ? 1.0f : -1.0f) /
                (fabsf(tau) + sqrtf(1.0f + tau * tau));
    float c = rsqrtf(1.0f + t * t);
    float s = t * c;
#pragma unroll
    for (int k = 0; k < 3; ++k) {
        float skp = S[k][p], skq = S[k][q];
        S[k][p] = c * skp - s * skq;
        S[k][q] = s * skp + c * skq;
    }
#pragma unroll
    for (int k = 0; k < 3; ++k) {
        float spk = S[p][k], sqk = S[q][k];
        S[p][k] = c * spk - s * sqk;
        S[q][k] = s * spk + c * sqk;
    }
#pragma unroll
    for (int k = 0; k < 3; ++k) {
        float vkp = V[k][p], vkq = V[k][q];
        V[k][p] = c * vkp - s * vkq;
        V[k][q] = s * vkp + c * vkq;
    }
}

__global__ void wsvd_stage2(const float* __restrict__ acc,
                            float* __restrict__ out, int B) {
    int b = blockIdx.x * blockDim.x + threadIdx.x;
    if (b >= B) return;
    const float* a = acc + b * 16;

    float ws  = a[15];
    float inv = 1.0f / fmaxf(ws, 1e-8f);
    float sw  = ws * inv;                    // sum of normalized weights (~1)
    float sb[3], tb[3];
#pragma unroll
    for (int c = 0; c < 3; ++c) { sb[c] = a[12 + c] * inv; tb[c] = a[9 + c] * inv; }

    // H = M - (2 - sw) * sb (x) tb   (exact expansion of centered covariance)
    float H[3][3];
#pragma unroll
    for (int i = 0; i < 3; ++i)
#pragma unroll
        for (int j = 0; j < 3; ++j)
            H[i][j] = a[i * 3 + j] * inv - (2.0f - sw) * sb[i] * tb[j];

    // S = H^T H, Jacobi eigendecomposition -> V, eigenvalues on diag
    float S[3][3];
#pragma unroll
    for (int i = 0; i < 3; ++i)
#pragma unroll
        for (int j = 0; j < 3; ++j)
            S[i][j] = H[0][i] * H[0][j] + H[1][i] * H[1][j] + H[2][i] * H[2][j];
    float V[3][3] = {{1, 0, 0}, {0, 1, 0}, {0, 0, 1}};
    for (int sweep = 0; sweep < 6; ++sweep) {
        jacobi_rot(S, V, 0, 1);
        jacobi_rot(S, V, 0, 2);
        jacobi_rot(S, V, 1, 2);
    }

    // sort eigenvalues descending (swap V columns along)
    float d[3] = {S[0][0], S[1][1], S[2][2]};
#define CSWAP(i, j)                                                     \
    if (d[i] < d[j]) {                                                  \
        float tmp = d[i]; d[i] = d[j]; d[j] = tmp;                      \
        for (int k = 0; k < 3; ++k) {                                   \
            float tv = V[k][i]; V[k][i] = V[k][j]; V[k][j] = tv;        \
        }                                                               \
    }
    CSWAP(0, 1) CSWAP(0, 2) CSWAP(1, 2)
#undef CSWAP

    // U via u_i = H v_i / sigma_i with Gram-Schmidt + cross-product fallback
    float u0[3], u1[3], u2[3];
#pragma unroll
    for (int i = 0; i < 3; ++i)
        u0[i] = H[i][0] * V[0][0] + H[i][1] * V[1][0] + H[i][2] * V[2][0];
    float n0 = sqrtf(u0[0] * u0[0] + u0[1] * u0[1] + u0[2] * u0[2]);
    if (n0 > 1e-12f) { float r = 1.0f / n0; u0[0] *= r; u0[1] *= r; u0[2] *= r; }
    else             { u0[0] = 1.0f; u0[1] = 0.0f; u0[2] = 0.0f; }

#pragma unroll
    for (int i = 0; i < 3; ++i)
        u1[i] = H[i][0] * V[0][1] + H[i][1] * V[1][1] + H[i][2] * V[2][1];
    float d01 = u0[0] * u1[0] + u0[1] * u1[1] + u0[2] * u1[2];
#pragma unroll
    for (int i = 0; i < 3; ++i) u1[i] -= d01 * u0[i];
    float n1 = sqrtf(u1[0] * u1[0] + u1[1] * u1[1] + u1[2] * u1[2]);
    if (n1 > 1e-12f) { float r = 1.0f / n1; u1[0] *= r; u1[1] *= r; u1[2] *= r; }
    else {
        // any unit vector orthogonal to u0
        float e[3] = {0.0f, 0.0f, 0.0f};
        int ax = (fabsf(u0[0]) <= fabsf(u0[1]) && fabsf(u0[0]) <= fabsf(u0[2])) ? 0
               : (fabsf(u0[1]) <= fabsf(u0[2]) ? 1 : 2);
        e[ax] = 1.0f;
        u1[0] = u0[1] * e[2] - u0[2] * e[1];
        u1[1] = u0[2] * e[0] - u0[0] * e[2];
        u1[2] = u0[0] * e[1] - u0[1] * e[0];
        float rn = rsqrtf(fmaxf(u1[0]*u1[0] + u1[1]*u1[1] + u1[2]*u1[2], 1e-30f));
        u1[0] *= rn; u1[1] *= rn; u1[2] *= rn;
    }
    u2[0] = u0[1] * u1[2] - u0[2] * u1[1];
    u2[1] = u0[2] * u1[0] - u0[0] * u1[2];
    u2[2] = u0[0] * u1[1] - u0[1] * u1[0];

    // reflection fix: with det(U)=+1 forced, det(R_raw)=det(V)
    float detV = V[0][0] * (V[1][1] * V[2][2] - V[1][2] * V[2][1])
               - V[0][1] * (V[1][0] * V[2][2] - V[1][2] * V[2][0])
               + V[0][2] * (V[1][0] * V[2][1] - V[1][1] * V[2][0]);
    float s3 = (detV >= 0.0f) ? 1.0f : -1.0f;

    float R[3][3];
#pragma unroll
    for (int i = 0; i < 3; ++i)
#pragma unroll
        for (int k = 0; k < 3; ++k)
            R[i][k] = V[i][0] * u0[k] + V[i][1] * u1[k] + s3 * V[i][2] * u2[k];

    // translation
    float t[3];
#pragma unroll
    for (int i = 0; i < 3; ++i)
        t[i] = tb[i] - (R[i][0] * sb[0] + R[i][1] * sb[1] + R[i][2] * sb[2]);

    // quaternion (Shepperd, scalar-last [x,y,z,w])
    float q[4];
    float trace = R[0][0] + R[1][1] + R[2][2];
    if (trace > 0.0f) {
        float s0 = sqrtf(fmaxf(trace + 1.0f, 1e-10f)) * 2.0f;
        q[0] = (R[2][1] - R[1][2]) / s0;
        q[1] = (R[0][2] - R[2][0]) / s0;
        q[2] = (R[1][0] - R[0][1]) / s0;
        q[3] = 0.25f * s0;
    } else if (R[0][0] > R[1][1] && R[0][0] > R[2][2]) {
        float s1 = sqrtf(fmaxf(1.0f + R[0][0] - R[1][1] - R[2][2], 1e-10f)) * 2.0f;
        q[0] = 0.25f * s1;
        q[1] = (R[0][1] + R[1][0]) / s1;
        q[2] = (R[0][2] + R[2][0]) / s1;
        q[3] = (R[2][1] - R[1][2]) / s1;
    } else if (R[1][1] > R[2][2]) {
        float s2 = sqrtf(fmaxf(1.0f + R[1][1] - R[0][0] - R[2][2], 1e-10f)) * 2.0f;
        q[0] = (R[0][1] + R[1][0]) / s2;
        q[1] = 0.25f * s2;
        q[2] = (R[1][2] + R[2][1]) / s2;
        q[3] = (R[0][2] - R[2][0]) / s2;
    } else {
        float sq = sqrtf(fmaxf(1.0f + R[2][2] - R[0][0] - R[1][1], 1e-10f)) * 2.0f;
        q[0] = (R[0][2] + R[2][0]) / sq;
        q[1] = (R[1][2] + R[2][1]) / sq;
        q[2] = 0.25f * sq;
        q[3] = (R[1][0] - R[0][1]) / sq;
    }
    float qn = rsqrtf(fmaxf(q[0]*q[0] + q[1]*q[1] + q[2]*q[2] + q[3]*q[3], 1e-30f));
#pragma unroll
    for (int i = 0; i < 4; ++i) q[i] *= qn;

    // outputs: q [B,4] | t [B,3] | R [B,3,3]
#pragma unroll
    for (int i = 0; i < 4; ++i) out[b * 4 + i] = q[i];
#pragma unroll
    for (int i = 0; i < 3; ++i) out[B * 4 + b * 3 + i] = t[i];
#pragma unroll
    for (int i = 0; i < 3; ++i)
#pragma unroll
        for (int j = 0; j < 3; ++j)
            out[B * 7 + b * 9 + i * 3 + j] = R[i][j];
}

// ---------------------------------------------------------------------------
extern "C" void kernel_launch(void* const* d_in, const int* in_sizes, int n_in,
                              void* d_out, int out_size, void* d_ws, size_t ws_size,
                              hipStream_t stream) {
    const float* corr    = (const float*)d_in[0];
    const float* overlap = (const float*)d_in[1];
    const float* src     = (const float*)d_in[2];
    const float* tgt     = (const float*)d_in[3];
    float* out = (float*)d_out;
    float* acc = (float*)d_ws;

    const int B = in_sizes[1] / L_DIM;  // overlap_scores is [B, L, 1]

    wsvd_zero<<<(B * 16 + 255) / 256, 256, 0, stream>>>(acc, B * 16);
    wsvd_stage1<<<B * TILE_GROUPS, 256, 0, stream>>>(corr, overlap, src, tgt, acc);
    wsvd_stage2<<<(B + 127) / 128, 128, 0, stream>>>(acc, out, B);
}